// ExpertRouter_68539088109737
// MI455X (gfx1250) — compile-verified
//
#include <hip/hip_runtime.h>
#include <math.h>

typedef float v2f __attribute__((ext_vector_type(2)));
typedef float v8f __attribute__((ext_vector_type(8)));

#define D_MODEL 4096
#define N_EXPERTS 64
#define TOKENS_PER_WAVE 16
#define WAVES_PER_BLOCK 4
#define LDS_STRIDE 65   // 64 experts + 1 pad -> conflict-free per-token row reads

// One wave computes a 16-token x 64-expert logit tile with V_WMMA_F32_16X16X4_F32
// (exact f32 math, matching the reference), then does softmax + top-2 per token.
// A-stream (hidden_states, 256MB, streamed once) is loaded non-temporally so the
// heavily-reused W (1MB) stays resident in WGP$/L2.
__global__ __launch_bounds__(WAVES_PER_BLOCK * 32)
void ExpertRouter_kernel(const float* __restrict__ hs,    // [B*S, D] flat tokens
                         const float* __restrict__ rb,    // [B, 64]
                         const float* __restrict__ W,     // [64, D]
                         const float* __restrict__ bias,  // [64]
                         float* __restrict__ out_w,       // [B*S, 2]
                         int*   __restrict__ out_i)       // [B*S, 2]
{
    __shared__ float lds[WAVES_PER_BLOCK * TOKENS_PER_WAVE * LDS_STRIDE];

    const int lane = threadIdx.x & 31;
    const int wave = threadIdx.x >> 5;
    const int t0   = (blockIdx.x * WAVES_PER_BLOCK + wave) * TOKENS_PER_WAVE;
    const int batch = t0 >> 12;            // token / 4096 (tile never crosses batch)

    const int half = lane >> 4;            // 0: K pair {k,k+1}, 1: {k+2,k+3}
    const int sub  = lane & 15;            // M (token row) for A, N (expert col) for B

    // A layout (32-bit 16x4): lane L holds M=L%16, K = k + 2*(L/16) + {0,1}
    const float* aPtr = hs + (size_t)(t0 + sub) * D_MODEL + 2 * half;
    // B layout (mirror): lane L holds N=L%16, K = k + 2*(L/16) + {0,1}; B[k][n] = W[n][k]
    const float* bPtr = W + (size_t)sub * D_MODEL + 2 * half;

    v8f acc[4];
    #pragma unroll
    for (int t = 0; t < 4; ++t) acc[t] = (v8f){0.f,0.f,0.f,0.f,0.f,0.f,0.f,0.f};

    // Unroll 8: ~36 outstanding b64 loads per wave (LOADcnt is 6-bit) to cover
    // HBM latency, since grid parallelism is only ~1 wave/SIMD.
    #pragma unroll 8
    for (int k = 0; k < D_MODEL; k += 4) {
        v2f a  = __builtin_nontemporal_load((const v2f*)(aPtr + k));  // stream-once: TH=NT
        v2f b0 = *(const v2f*)(bPtr + k);
        v2f b1 = *(const v2f*)(bPtr + 16 * D_MODEL + k);
        v2f b2 = *(const v2f*)(bPtr + 32 * D_MODEL + k);
        v2f b3 = *(const v2f*)(bPtr + 48 * D_MODEL + k);
        acc[0] = __builtin_amdgcn_wmma_f32_16x16x4_f32(false, a, false, b0, (short)0, acc[0], false, false);
        acc[1] = __builtin_amdgcn_wmma_f32_16x16x4_f32(false, a, false, b1, (short)0, acc[1], false, false);
        acc[2] = __builtin_amdgcn_wmma_f32_16x16x4_f32(false, a, false, b2, (short)0, acc[2], false, false);
        acc[3] = __builtin_amdgcn_wmma_f32_16x16x4_f32(false, a, false, b3, (short)0, acc[3], false, false);
    }

    // Add linear bias + routing bias; expert column is constant per lane per tile.
    float* myLds = lds + wave * TOKENS_PER_WAVE * LDS_STRIDE;
    #pragma unroll
    for (int t = 0; t < 4; ++t) {
        const int e = t * 16 + sub;
        const float bv = bias[e] + rb[batch * N_EXPERTS + e];
        #pragma unroll
        for (int r = 0; r < 8; ++r) {
            const int m = r + 8 * half;    // C layout: VGPR r -> M = r (+8 for hi lanes)
            myLds[m * LDS_STRIDE + e] = acc[t][r] + bv;
        }
    }
    __syncthreads();

    // Lanes 0..15: one token each -> softmax over 64, top-2, renormalize.
    if (lane < 16) {
        const float* row = myLds + lane * LDS_STRIDE;
        float mx = row[0];
        #pragma unroll
        for (int e = 1; e < N_EXPERTS; ++e) mx = fmaxf(mx, row[e]);

        float Z = 0.f;
        float v1 = -INFINITY, v2 = -INFINITY;
        int   i1 = 0, i2 = 0;
        #pragma unroll
        for (int e = 0; e < N_EXPERTS; ++e) {
            const float l = row[e];
            Z += expf(l - mx);
            if (l > v1) { v2 = v1; i2 = i1; v1 = l; i1 = e; }
            else if (l > v2) { v2 = l; i2 = e; }
        }
        const float s1 = expf(v1 - mx) / Z;
        const float s2 = expf(v2 - mx) / Z;
        const float inv = 1.0f / (s1 + s2 + 1e-8f);

        const int tok = t0 + lane;
        out_w[tok * 2 + 0] = s1 * inv;
        out_w[tok * 2 + 1] = s2 * inv;
        out_i[tok * 2 + 0] = i1;
        out_i[tok * 2 + 1] = i2;
    }
}

extern "C" void kernel_launch(void* const* d_in, const int* in_sizes, int n_in,
                              void* d_out, int out_size, void* d_ws, size_t ws_size,
                              hipStream_t stream) {
    const float* hs   = (const float*)d_in[0];   // [4, 4096, 4096]
    const float* rb   = (const float*)d_in[1];   // [4, 64]
    const float* W    = (const float*)d_in[2];   // [64, 4096]
    const float* bias = (const float*)d_in[3];   // [64]

    const int total_tokens = 4 * 4096;
    float* out_w = (float*)d_out;                          // [tokens, 2] f32
    int*   out_i = (int*)((float*)d_out + total_tokens*2); // [tokens, 2] i32

    const int blocks = total_tokens / (WAVES_PER_BLOCK * TOKENS_PER_WAVE); // 256
    ExpertRouter_kernel<<<blocks, WAVES_PER_BLOCK * 32, 0, stream>>>(
        hs, rb, W, bias, out_w, out_i);
}